// LFQ_40690520163080
// MI455X (gfx1250) — compile-verified
//
#include <hip/hip_runtime.h>
#include <cstdint>
#include <cstddef>

// ---------------------------------------------------------------------------
// LFQ fused kernel for MI455X (gfx1250, wave32, WMMA + async/TDM)
//   w_in stage : TENSOR_LOAD_TO_LDS (TDM, D# with LDS padding), s_wait_tensorcnt
//   proj_in    : x K-chunks staged via GLOBAL_LOAD_ASYNC_TO_LDS_B128,
//                double-buffered on ASYNCcnt; V_WMMA_F32_16X16X4_F32 chain
//   quantize   : sign -> ±1, 12-bit index pack
//   entropy    : factorized softmax over 4096 codes (12 Bernoullis), LDS accum
//   proj_out   : V_WMMA_F32_16X16X4_F32, K=12 (3 steps), N=512 (32 tiles)
// ---------------------------------------------------------------------------

#define DIM   512
#define CD    12
#define CS    4096
#define TPB   64          // tokens per block (4 waves x 16)
#define NTHR  128
#define EPSF  1e-5f
#define KC    32          // K elements staged per async chunk
#define NCH   (DIM / KC)  // 16 chunks

typedef float v2f __attribute__((ext_vector_type(2)));
typedef float v8f __attribute__((ext_vector_type(8)));
typedef unsigned int u32x4 __attribute__((ext_vector_type(4)));
typedef int i32x4 __attribute__((ext_vector_type(4)));
typedef int i32x8 __attribute__((ext_vector_type(8)));

// LDS layout (floats). Total 14912 floats = 59,648 B (< 64 KB).
#define WSZ_OFF  0                    // w_in staged: 16 rows, stride 516
                                      //   TDM writes rows 0..11 (512 cols + 2x2 pad)
#define WSZ_SZ   (16 * 516)           // 8256
#define XBUF_OFF (WSZ_OFF + WSZ_SZ)   // x staging: 4 waves * 2 buf * (16*36) = 4608
#define AVG_OFF  XBUF_OFF             // ALIAS: avg_prob[4096] (used after proj_in)
#define UNION_SZ 4608
#define XP_OFF   (XBUF_OFF + UNION_SZ) // 12864: xp tiles, 4 waves * [16][16]
#define Q_OFF    (XP_OFF + 4 * 256)    // 13888: quantized tiles
#define SM_FLOATS (Q_OFF + 4 * 256)    // 14912

__global__ void lfq_zero(float* __restrict__ g) {
  int i = blockIdx.x * 256 + threadIdx.x;
  if (i < CS + 1) g[i] = 0.f;
}

__global__ __launch_bounds__(NTHR) void lfq_fused(
    const float* __restrict__ x,
    const float* __restrict__ w_in,
    const float* __restrict__ b_in,
    const float* __restrict__ w_out,
    const float* __restrict__ b_out,
    float* __restrict__ out,
    int*   __restrict__ indices,
    float* __restrict__ gacc)
{
  extern __shared__ float sm[];
  const int tid  = threadIdx.x;
  const int lane = tid & 31;
  const int wave = tid >> 5;
  const int half = lane >> 4;    // WMMA half-wave (K-pair select)
  const int rn   = lane & 15;    // A-matrix row (token) / B,D column
  const int kks  = 2 * half;     // K offset within a 4-wide WMMA step

  // ---- zero w_in pad rows 12..15 (TDM fills rows 0..11) ----
  for (int i = tid; i < 4 * 516; i += NTHR) sm[WSZ_OFF + 12 * 516 + i] = 0.f;

  // ---- TDM: tensor_load_to_lds of w_in [12][512] f32 with LDS padding:
  //      pad_interval=256 dwords, pad_amount=2 dwords -> row stride 516 ----
  if (wave == 0) {
    const uint32_t lds = (uint32_t)(uintptr_t)(sm + WSZ_OFF);
    const uint64_t ga  = (uint64_t)(uintptr_t)w_in;
    u32x4 g0;
    g0[0] = 1u;                                              // count=1 (valid D#)
    g0[1] = lds;                                             // lds_addr
    g0[2] = (uint32_t)ga;                                    // global_addr lo
    g0[3] = (uint32_t)((ga >> 32) & 0x01FFFFFFu) | (2u << 30); // addr hi | type=2
    i32x8 g1;
    g1[0] = (2 << 16) | (1 << 20) | (7 << 22) | (1 << 25);   // 4B elems, pad_en,
                                                             // interval=256dw, amt=2dw
    g1[1] = (int)(512u << 16);   // tensor_dim0 = 512 (bits 79:48)
    g1[2] = (int)(12u << 16);    // tensor_dim1 = 12  (bits 111:80)
    g1[3] = (int)(512u << 16);   // tile_dim0 = 512   (bits 127:112)
    g1[4] = 12;                  // tile_dim1 = 12, tile_dim2 = 0
    g1[5] = 512;                 // tensor_dim0_stride = 512
    g1[6] = 0; g1[7] = 0;
    i32x4 gz = {};
#if __clang_major__ >= 23
    i32x8 gz8 = {};
    __builtin_amdgcn_tensor_load_to_lds(g0, g1, gz, gz, gz8, 0);
#else
    __builtin_amdgcn_tensor_load_to_lds(g0, g1, gz, gz, 0);
#endif
    __builtin_amdgcn_s_wait_tensorcnt(0);
  }
  __syncthreads();   // wsz ready for all waves

  const int tokBase = blockIdx.x * TPB + wave * 16;
  float* xpw = sm + XP_OFF + wave * 256;   // [16 tokens][16]
  float* qw  = sm + Q_OFF  + wave * 256;   // [16 tokens][16]
  float* xw  = sm + XBUF_OFF + wave * 1152;              // 2 x [16][36]
  const uint32_t xwB = (uint32_t)(uintptr_t)xw;

  // ---- proj_in: xp[16x12] = x_tile[16x512] @ w_in^T, WMMA f32 16x16x4.
  //      x staged in KC=32 chunks via async B128 loads, double-buffered. ----
  // prologue: stage chunks 0 and 1
  #pragma unroll
  for (int c = 0; c < 2; ++c) {
    #pragma unroll
    for (int q = 0; q < 4; ++q) {
      const int e = q * 32 + lane;            // 16B element id within chunk
      const int row = e >> 3, col = (e & 7) * 4;
      const uint64_t ga = (uint64_t)(uintptr_t)(x + (size_t)(tokBase + row) * DIM + c * KC + col);
      const uint32_t la = xwB + (uint32_t)(c * 576 + row * 36 + col) * 4u;
      asm volatile("global_load_async_to_lds_b128 %0, %1, off"
                   :: "v"(la), "v"(ga) : "memory");
    }
  }

  v8f acc = {};
  for (int c = 0; c < NCH; ++c) {
    if (c < NCH - 1) asm volatile("s_wait_asynccnt 4" ::: "memory");
    else             asm volatile("s_wait_asynccnt 0" ::: "memory");
    const int buf   = (c & 1) * 576;
    const int shift = (c >= 8) ? 2 : 0;       // skip LDS pad dwords in wsz rows
    const float* ap = xw + buf + rn * 36 + kks;
    const float* bp = sm + WSZ_OFF + rn * 516 + c * KC + kks + shift;
    #pragma unroll
    for (int s = 0; s < KC / 4; ++s) {
      v2f a = *(const v2f*)(ap + s * 4);
      v2f b = *(const v2f*)(bp + s * 4);
      acc = __builtin_amdgcn_wmma_f32_16x16x4_f32(false, a, false, b,
                                                  (short)0, acc, false, false);
    }
    if (c + 2 < NCH) {                        // refill the buffer just consumed
      #pragma unroll
      for (int q = 0; q < 4; ++q) {
        const int e = q * 32 + lane;
        const int row = e >> 3, col = (e & 7) * 4;
        const uint64_t ga = (uint64_t)(uintptr_t)(x + (size_t)(tokBase + row) * DIM + (c + 2) * KC + col);
        const uint32_t la = xwB + (uint32_t)(buf + row * 36 + col) * 4u;
        asm volatile("global_load_async_to_lds_b128 %0, %1, off"
                     :: "v"(la), "v"(ga) : "memory");
      }
    }
  }
  {
    float bias = (rn < CD) ? b_in[rn] : 0.f;
    #pragma unroll
    for (int v = 0; v < 8; ++v)
      xpw[(v + 8 * half) * 16 + rn] = acc[v] + bias;  // D: lane=(col), vgpr=(row)
  }
  __syncthreads();   // xp visible; ALL waves done with xbuf region

  // ---- zero block-local avg_prob (aliases xbuf) + quantize / pack indices ----
  for (int i = tid; i < CS; i += NTHR) sm[AVG_OFF + i] = 0.f;
  if (lane < 16) {
    int idx = 0;
    #pragma unroll
    for (int j = 0; j < CD; ++j) {
      float v = xpw[lane * 16 + j];
      int bit = v > 0.f;
      idx |= bit << (11 - j);
      qw[lane * 16 + j] = bit ? 1.f : -1.f;
    }
    #pragma unroll
    for (int j = CD; j < 16; ++j) qw[lane * 16 + j] = 0.f;  // K-pad
    indices[tokBase + lane] = idx;
  }
  __syncthreads();

  // ---- proj_out: out[16x512] = q[16x12] @ w_out^T + b_out, 3 K-steps ----
  v2f a0 = *(const v2f*)(qw + rn * 16 + 0 + kks);
  v2f a1 = *(const v2f*)(qw + rn * 16 + 4 + kks);
  v2f a2 = *(const v2f*)(qw + rn * 16 + 8 + kks);
  for (int dt = 0; dt < 32; ++dt) {
    const int dbase = dt * 16;
    const float bias = b_out[dbase + rn];
    v8f d8;
    #pragma unroll
    for (int v = 0; v < 8; ++v) d8[v] = bias;
    const float* wrow = w_out + (size_t)(dbase + rn) * CD + kks;
    v2f b0, b1, b2;
    b0.x = wrow[0]; b0.y = wrow[1];
    b1.x = wrow[4]; b1.y = wrow[5];
    b2.x = wrow[8]; b2.y = wrow[9];
    d8 = __builtin_amdgcn_wmma_f32_16x16x4_f32(false, a0, false, b0, (short)0, d8, false, false);
    d8 = __builtin_amdgcn_wmma_f32_16x16x4_f32(false, a1, false, b1, (short)0, d8, false, false);
    d8 = __builtin_amdgcn_wmma_f32_16x16x4_f32(false, a2, false, b2, (short)0, d8, false, false);
    #pragma unroll
    for (int v = 0; v < 8; ++v)
      out[(size_t)(tokBase + v + 8 * half) * DIM + dbase + rn] = d8[v];
  }

  // ---- entropy: softmax over 4096 codes factorizes into 12 Bernoullis.
  //      prob(c) = hi[c>>6] * lo[c&63]; lane covers codes c = i*32 + lane
  //      (bank-conflict-free LDS accumulation). ----
  float ent = 0.f;
  for (int t = 0; t < 16; ++t) {
    float p[CD];
    #pragma unroll
    for (int j = 0; j < CD; ++j) {
      float xv = xpw[t * 16 + j];                 // LDS broadcast read
      p[j] = 1.f / (1.f + __expf(-400.f * xv));   // sigmoid(2*inv_temp*2*x)
    }
    // hi table over code bits 11..6 (dims j = 5-b)
    float hi[64];
    hi[0] = 1.f;
    #pragma unroll
    for (int b = 0; b < 6; ++b) {
      const int j = 5 - b;
      const int len = 1 << b;
      #pragma unroll
      for (int k = 0; k < len; ++k) {
        hi[k + len] = hi[k] * p[j];
        hi[k]       = hi[k] * (1.f - p[j]);
      }
    }
    // low 6 bits of this lane's codes: bits0..4 = lane bits, bit5 = (i&1)
    float base = 1.f;
    #pragma unroll
    for (int b = 0; b < 5; ++b)
      base *= ((lane >> b) & 1) ? p[11 - b] : (1.f - p[11 - b]);
    const float lo0 = base * (1.f - p[6]);
    const float lo1 = base * p[6];
    #pragma unroll
    for (int i = 0; i < 128; ++i) {
      float prob = hi[i >> 1] * ((i & 1) ? lo1 : lo0);
      __hip_atomic_fetch_add(&sm[AVG_OFF + i * 32 + lane], prob,
                             __ATOMIC_RELAXED, __HIP_MEMORY_SCOPE_WORKGROUP);
      ent -= prob * __logf(fmaxf(prob, EPSF));
    }
  }
  // wave-reduce per-sample entropy, one global atomic per wave
  #pragma unroll
  for (int off = 16; off > 0; off >>= 1)
    ent += __shfl_down(ent, off);
  if (lane == 0)
    __hip_atomic_fetch_add(&gacc[CS], ent, __ATOMIC_RELAXED,
                           __HIP_MEMORY_SCOPE_AGENT);

  // flush block-local avg_prob to global accumulator
  __syncthreads();
  for (int i = tid; i < CS; i += NTHR)
    __hip_atomic_fetch_add(&gacc[i], sm[AVG_OFF + i],
                           __ATOMIC_RELAXED, __HIP_MEMORY_SCOPE_AGENT);
}

__global__ void lfq_finish(const float* __restrict__ gacc, int T,
                           float* __restrict__ aux_out) {
  __shared__ float red[256];
  const float invT = 1.f / (float)T;
  float ce = 0.f;
  for (int i = threadIdx.x; i < CS; i += 256) {
    float ap = gacc[i] * invT;
    ce -= ap * __logf(fmaxf(ap, EPSF));
  }
  red[threadIdx.x] = ce;
  __syncthreads();
  for (int s = 128; s > 0; s >>= 1) {
    if (threadIdx.x < s) red[threadIdx.x] += red[threadIdx.x + s];
    __syncthreads();
  }
  if (threadIdx.x == 0) {
    float pse = gacc[CS] * invT;                 // mean per-sample entropy
    aux_out[0] = 0.1f * (pse - 1.0f * red[0]);   // weight*(pse - gamma*ce)
  }
}

extern "C" void kernel_launch(void* const* d_in, const int* in_sizes, int n_in,
                              void* d_out, int out_size, void* d_ws, size_t ws_size,
                              hipStream_t stream) {
  const float* x     = (const float*)d_in[0];
  const float* w_in  = (const float*)d_in[1];
  const float* b_in  = (const float*)d_in[2];
  const float* w_out = (const float*)d_in[3];
  const float* b_out = (const float*)d_in[4];
  const int T = in_sizes[0] / DIM;               // 16384 tokens

  float* fout = (float*)d_out;                   // out: T*512 floats
  int*   iout = (int*)(fout + (size_t)T * DIM);  // indices: T int32 (bit-cast slot)
  float* aux  = fout + (size_t)T * DIM + T;      // aux_loss scalar
  float* gacc = (float*)d_ws;                    // [4096] avg_prob sums + [1] ent sum

  lfq_zero<<<(CS + 1 + 255) / 256, 256, 0, stream>>>(gacc);
  lfq_fused<<<T / TPB, NTHR, SM_FLOATS * sizeof(float), stream>>>(
      x, w_in, b_in, w_out, b_out, fout, iout, gacc);
  lfq_finish<<<1, 256, 0, stream>>>(gacc, T, aux);
}